// SimpleGCNEncoder_11012296147170
// MI455X (gfx1250) — compile-verified
//
#include <hip/hip_runtime.h>

#define D 128
#define EPSV 1e-5f

typedef __attribute__((ext_vector_type(2))) float v2f;
typedef __attribute__((ext_vector_type(8))) float v8f;

// ---------------------------------------------------------------------------
// Y[n x 128] = act(H) @ W + bias, act = optional fused BN+ReLU from previous
// layer (scale==nullptr -> identity).
// Block: 256 threads = 8 waves, computes a 64-row x 128-col output slab.
// Wave w owns columns [16w,16w+16) and keeps 4 accumulators (rows m0+16t).
// Each B fragment (from L2-hot W) feeds 4 independent v_wmma_f32_16x16x4_f32,
// amortizing W traffic 4x and giving 4 parallel WMMA dependency chains.
// ---------------------------------------------------------------------------
__global__ __launch_bounds__(256)
void gcn_gemm_bn_bias(const float* __restrict__ H,
                      const float* __restrict__ scale,   // may be null
                      const float* __restrict__ shift,   // may be null
                      const float* __restrict__ W,
                      const float* __restrict__ bias,
                      float* __restrict__ Y, int n)
{
    __shared__ float hs[64][132];  // pad 128->132: conflict-free column reads

    const int m0 = blockIdx.x << 6;

    // Stage the 64x128 H tile with float4 loads, fusing BN+ReLU.
    for (int i = threadIdx.x; i < 64 * (D / 4); i += 256) {
        int r = i >> 5;                 // 32 float4 per row
        int c = (i & 31) << 2;
        int gr = m0 + r; if (gr >= n) gr = n - 1;   // clamp for tail block
        float4 v = *(const float4*)(H + (size_t)gr * D + c);
        if (scale) {
            v.x = fmaxf(fmaf(v.x, scale[c + 0], shift[c + 0]), 0.0f);
            v.y = fmaxf(fmaf(v.y, scale[c + 1], shift[c + 1]), 0.0f);
            v.z = fmaxf(fmaf(v.z, scale[c + 2], shift[c + 2]), 0.0f);
            v.w = fmaxf(fmaf(v.w, scale[c + 3], shift[c + 3]), 0.0f);
        }
        *(float4*)&hs[r][c] = v;
    }
    __syncthreads();

    const int lane = threadIdx.x & 31;
    const int wave = threadIdx.x >> 5;
    const int mrow = lane & 15;                  // A: M index = lane % 16
    const int kh   = (lane >> 4) << 1;           // A/B: K sub-offset 0 or 2
    const int ncol = (wave << 4) + (lane & 15);  // B/C/D: N = lane % 16

    v8f acc0 = {}, acc1 = {}, acc2 = {}, acc3 = {};
    #pragma unroll 2
    for (int k = 0; k < D; k += 4) {
        v2f b;
        b.x = W[(size_t)(k + kh) * D + ncol];
        b.y = W[(size_t)(k + kh + 1) * D + ncol];
        v2f a0, a1, a2, a3;
        a0.x = hs[mrow     ][k + kh]; a0.y = hs[mrow     ][k + kh + 1];
        a1.x = hs[mrow + 16][k + kh]; a1.y = hs[mrow + 16][k + kh + 1];
        a2.x = hs[mrow + 32][k + kh]; a2.y = hs[mrow + 32][k + kh + 1];
        a3.x = hs[mrow + 48][k + kh]; a3.y = hs[mrow + 48][k + kh + 1];
        acc0 = __builtin_amdgcn_wmma_f32_16x16x4_f32(false, a0, false, b, (short)0, acc0, false, false);
        acc1 = __builtin_amdgcn_wmma_f32_16x16x4_f32(false, a1, false, b, (short)0, acc1, false, false);
        acc2 = __builtin_amdgcn_wmma_f32_16x16x4_f32(false, a2, false, b, (short)0, acc2, false, false);
        acc3 = __builtin_amdgcn_wmma_f32_16x16x4_f32(false, a3, false, b, (short)0, acc3, false, false);
    }

    // Epilogue: one uniform full/partial branch per block; full path is a
    // clause of 8 global_store_b32 at immediate offsets (v*512B) per tile.
    const float bb  = bias[ncol];
    const int rsub  = (lane >> 4) << 3;          // C/D: row = v + 8*(lane/16)
    const bool full = (m0 + 64) <= n;

#define STORE_TILE(T, A)                                                      \
    {                                                                         \
        const int rb = m0 + ((T) << 4) + rsub;                                \
        float* p = Y + (size_t)rb * D + ncol;                                 \
        if (full) {                                                           \
            _Pragma("unroll")                                                 \
            for (int v = 0; v < 8; ++v) p[(size_t)v * D] = (A)[v] + bb;       \
        } else {                                                              \
            _Pragma("unroll")                                                 \
            for (int v = 0; v < 8; ++v)                                       \
                if (rb + v < n) p[(size_t)v * D] = (A)[v] + bb;               \
        }                                                                     \
    }
    STORE_TILE(0, acc0)
    STORE_TILE(1, acc1)
    STORE_TILE(2, acc2)
    STORE_TILE(3, acc3)
#undef STORE_TILE
}

// ---------------------------------------------------------------------------
// COO SpMM scatter: Out[row[e], :] += w[e] * Hin[col[e], :]
// Block of 256 = 8 edges x 32 lanes; each lane handles 4 contiguous features
// (global_load_b128 gather + 4 hardware fp32 atomics). Activations are
// L2-resident (51 MB << 192 MB L2), so the irregular traffic stays on-chip.
// ---------------------------------------------------------------------------
__global__ __launch_bounds__(256)
void gcn_spmm_scatter(const float* __restrict__ ew,
                      const int* __restrict__ row,
                      const int* __restrict__ col,
                      const float* __restrict__ Hin,
                      float* __restrict__ Out, int E)
{
    const int lane = threadIdx.x & 31;
    const int e = blockIdx.x * 8 + (threadIdx.x >> 5);
    if (e >= E) return;

    const float w = ew[e];
    const int r = row[e], c = col[e];

    const float4 h = ((const float4*)(Hin + (size_t)c * D))[lane];
    float* outp = Out + (size_t)r * D + lane * 4;
    unsafeAtomicAdd(outp + 0, w * h.x);
    unsafeAtomicAdd(outp + 1, w * h.y);
    unsafeAtomicAdd(outp + 2, w * h.z);
    unsafeAtomicAdd(outp + 3, w * h.w);
}

// ---------------------------------------------------------------------------
// BN stats, phase 1: per-column partial sum / sum-of-squares into L2 atomics.
// Block of 256 = 2 rows x 128 columns; grid-strided over rows.
// ---------------------------------------------------------------------------
__global__ __launch_bounds__(256)
void gcn_bn_stats(const float* __restrict__ H,
                  float* __restrict__ sums, float* __restrict__ sumsq, int n)
{
    const int f  = threadIdx.x & (D - 1);
    const int r0 = threadIdx.x >> 7;   // 0..1
    float s = 0.0f, s2 = 0.0f;
    for (int r = blockIdx.x * 2 + r0; r < n; r += gridDim.x * 2) {
        float v = H[(size_t)r * D + f];
        s += v;
        s2 = fmaf(v, v, s2);
    }
    __shared__ float sh[256], sh2[256];
    sh[threadIdx.x] = s; sh2[threadIdx.x] = s2;
    __syncthreads();
    if (threadIdx.x < D) {
        unsafeAtomicAdd(&sums[f],  sh[f]  + sh[f + D]);
        unsafeAtomicAdd(&sumsq[f], sh2[f] + sh2[f + D]);
    }
}

// ---------------------------------------------------------------------------
// BN stats, phase 2: scale = gamma * rsqrt(var + eps), shift = beta - mean*scale
// (biased variance, matching h.var(axis=0)). One block of 128 threads.
// ---------------------------------------------------------------------------
__global__ void gcn_bn_finalize(const float* __restrict__ sums,
                                const float* __restrict__ sumsq,
                                const float* __restrict__ g,
                                const float* __restrict__ be,
                                float* __restrict__ scale,
                                float* __restrict__ shift, int n)
{
    const int f = threadIdx.x;
    const float inv_n = 1.0f / (float)n;
    const float m   = sums[f] * inv_n;
    const float var = fmaf(-m, m, sumsq[f] * inv_n);
    const float sc  = g[f] * rsqrtf(var + EPSV);
    scale[f] = sc;
    shift[f] = fmaf(-m, sc, be[f]);
}

// ---------------------------------------------------------------------------
extern "C" void kernel_launch(void* const* d_in, const int* in_sizes, int n_in,
                              void* d_out, int out_size, void* d_ws, size_t ws_size,
                              hipStream_t stream)
{
    const float* x   = (const float*)d_in[0];
    const float* ew  = (const float*)d_in[1];
    const float* W0  = (const float*)d_in[2];
    const float* b0  = (const float*)d_in[3];
    const float* g0  = (const float*)d_in[4];
    const float* be0 = (const float*)d_in[5];
    const float* W1  = (const float*)d_in[6];
    const float* b1  = (const float*)d_in[7];
    const float* g1  = (const float*)d_in[8];
    const float* be1 = (const float*)d_in[9];
    const float* W2  = (const float*)d_in[10];
    const float* b2  = (const float*)d_in[11];
    const int*   row = (const int*)d_in[12];
    const int*   col = (const int*)d_in[13];

    const int n = in_sizes[0] / D;     // 100000
    const int E = in_sizes[1];         // 1600000
    float* out = (float*)d_out;

    // Workspace: two N x 128 fp32 ping-pong buffers + BN scratch.
    char* ws = (char*)d_ws;
    const size_t ndb    = (size_t)n * D * sizeof(float);
    const size_t ndb_al = (ndb + 255) & ~(size_t)255;
    float* bufA  = (float*)ws;
    float* bufB  = (float*)(ws + ndb_al);
    float* sums  = (float*)(ws + 2 * ndb_al);
    float* sumsq = sums + D;
    float* scale = sums + 2 * D;
    float* shift = sums + 3 * D;

    const int gemmGrid  = (n + 63) / 64;
    const int spmmGrid  = (E + 7) / 8;
    const int statsGrid = 512;

    // ---- layer 1: h = relu(BN(A @ (x W0 + b0))) ----
    gcn_gemm_bn_bias<<<gemmGrid, 256, 0, stream>>>(x, nullptr, nullptr, W0, b0, bufA, n);
    hipMemsetAsync(bufB, 0, ndb, stream);
    gcn_spmm_scatter<<<spmmGrid, 256, 0, stream>>>(ew, row, col, bufA, bufB, E);
    hipMemsetAsync(sums, 0, 2 * D * sizeof(float), stream);
    gcn_bn_stats<<<statsGrid, 256, 0, stream>>>(bufB, sums, sumsq, n);
    gcn_bn_finalize<<<1, D, 0, stream>>>(sums, sumsq, g0, be0, scale, shift, n);

    // ---- layer 2 (BN+ReLU of layer-1 fused into this GEMM's tile load) ----
    gcn_gemm_bn_bias<<<gemmGrid, 256, 0, stream>>>(bufB, scale, shift, W1, b1, bufA, n);
    hipMemsetAsync(bufB, 0, ndb, stream);
    gcn_spmm_scatter<<<spmmGrid, 256, 0, stream>>>(ew, row, col, bufA, bufB, E);
    hipMemsetAsync(sums, 0, 2 * D * sizeof(float), stream);
    gcn_bn_stats<<<statsGrid, 256, 0, stream>>>(bufB, sums, sumsq, n);
    gcn_bn_finalize<<<1, D, 0, stream>>>(sums, sumsq, g1, be1, scale, shift, n);

    // ---- layer 3: out = A @ (relu(BN(h)) W2 + b2) ----
    gcn_gemm_bn_bias<<<gemmGrid, 256, 0, stream>>>(bufB, scale, shift, W2, b2, bufA, n);
    hipMemsetAsync(out, 0, (size_t)out_size * sizeof(float), stream);
    gcn_spmm_scatter<<<spmmGrid, 256, 0, stream>>>(ew, row, col, bufA, out, E);
}